// BasicBlock_88665304858673
// MI455X (gfx1250) — compile-verified
//
#include <hip/hip_runtime.h>
#include <hip/hip_bf16.h>

typedef __attribute__((ext_vector_type(2))) float v2f;
typedef __attribute__((ext_vector_type(8))) float v8f;

#define BM 128
#define BN 128
#define BK 16
#define LDA 20   // padded LDS row stride (floats), even -> float2 aligned, spreads banks

// ---------------------------------------------------------------------------
// Generic implicit-GEMM conv:  Out[b,co,ho,wo] = scale[co]*conv(X,W) + bias[co]
// M = Cout (512), N = B*Ho*Wo (6272), K = Cin*ks*ks (mult of 16)
// grid = (Cout/BM, N/BN), block = 256 threads (8 waves)
// ---------------------------------------------------------------------------
__global__ __launch_bounds__(256)
void conv_wmma_kernel(const float* __restrict__ X, const float* __restrict__ W,
                      const float* __restrict__ scale, const float* __restrict__ bias,
                      float* __restrict__ Out,
                      int Cin, int Hin, int Win, int ksize, int stride, int pad,
                      int Cout, int Ho, int Wo, int Batch)
{
    const int K     = Cin * ksize * ksize;
    const int ks2   = ksize * ksize;
    const int HoWo  = Ho * Wo;
    const int HinWin = Hin * Win;

    __shared__ float ldsA[BM * LDA];   // [m][k] weights tile
    __shared__ float ldsB[BN * LDA];   // [n][k] im2col tile (transposed for b64 reads)

    const int tid  = threadIdx.x;
    const int wave = tid >> 5;
    const int lane = tid & 31;
    const int block_m = blockIdx.x * BM;
    const int block_n = blockIdx.y * BN;

    const int wtile_m = (wave >> 1) * 32;   // 4 waves along M: 32 rows each
    const int wtile_n = (wave & 1) * 64;    // 2 waves along N: 64 cols each

    v8f acc[2][4];
#pragma unroll
    for (int i = 0; i < 2; ++i)
#pragma unroll
        for (int j = 0; j < 4; ++j) acc[i][j] = (v8f)0.f;

    const int l_m  = lane & 15;            // M (or N) index within 16
    const int l_kq = (lane >> 4) * 2;      // K sub-pair select: 0 or 2

    // ---- precompute per-thread B-staging constants (hoisted out of K loop) ----
    // thread stages elements (nl, kl): nl = (tid>>4) + 16*j (j=0..7), kl = tid&15
    const int kl  = tid & 15;
    const int nl0 = tid >> 4;
    int nbase[8], hbase[8], wbase[8];
#pragma unroll
    for (int j = 0; j < 8; ++j) {
        int n  = block_n + nl0 + 16 * j;
        int b  = n / HoWo;
        int p  = n - b * HoWo;
        int ho = p / Wo;
        int wo = p - ho * Wo;
        nbase[j] = b * Cin * HinWin;
        hbase[j] = ho * stride - pad;
        wbase[j] = wo * stride - pad;
    }
    // incremental (cin, r) tracker for kk = kb + kl
    int cin = kl / ks2;
    int rr  = kl - cin * ks2;

    for (int kb = 0; kb < K; kb += BK) {
        // ---- stage A (weights, row-major M x K, contiguous in k) ----
#pragma unroll
        for (int v = tid; v < (BM * BK) / 4; v += 256) {
            int m  = v >> 2;           // 4 float4 per 16-wide row
            int kq = (v & 3) * 4;
            const float4 d = *(const float4*)(W + (size_t)(block_m + m) * K + kb + kq);
            *(float4*)&ldsA[m * LDA + kq] = d;
        }
        // prefetch next weight tile into L2 (global_prefetch_b8 path)
        if (kb + BK < K)
            __builtin_prefetch(W + (size_t)(block_m + (tid >> 1)) * K + kb + BK, 0, 1);

        // ---- stage B (im2col activations, division-free inner loop) ----
        {
            const int kh = rr / ksize;
            const int kw = rr - kh * ksize;
            const int coff = cin * HinWin;
#pragma unroll
            for (int j = 0; j < 8; ++j) {
                int h = hbase[j] + kh;
                int w = wbase[j] + kw;
                float val = 0.f;
                if (h >= 0 && h < Hin && w >= 0 && w < Win)
                    val = X[(size_t)nbase[j] + coff + h * Win + w];
                ldsB[(nl0 + 16 * j) * LDA + kl] = val;
            }
            // advance k by BK for next tile: one small division per tile
            rr += BK;
            int q = rr / ks2;
            cin += q;
            rr  -= q * ks2;
        }
        __syncthreads();

        // ---- compute: 4 k-substeps of WMMA 16x16x4 f32 ----
#pragma unroll
        for (int k4 = 0; k4 < 4; ++k4) {
            const int koff = k4 * 4 + l_kq;
            v2f a[2], bf[4];
            a[0] = *(const v2f*)&ldsA[(wtile_m +      l_m) * LDA + koff];
            a[1] = *(const v2f*)&ldsA[(wtile_m + 16 + l_m) * LDA + koff];
#pragma unroll
            for (int j = 0; j < 4; ++j)
                bf[j] = *(const v2f*)&ldsB[(wtile_n + j * 16 + l_m) * LDA + koff];
#pragma unroll
            for (int i = 0; i < 2; ++i)
#pragma unroll
                for (int j = 0; j < 4; ++j)
                    acc[i][j] = __builtin_amdgcn_wmma_f32_16x16x4_f32(
                        false, a[i], false, bf[j], (short)0, acc[i][j], false, false);
        }
        __syncthreads();
    }

    // ---- epilogue: scale/bias (folded BN), scatter to NCHW ----
#pragma unroll
    for (int i = 0; i < 2; ++i) {
#pragma unroll
        for (int j = 0; j < 4; ++j) {
            const int base_m = block_m + wtile_m + i * 16 + (lane >> 4) * 8;
            const int n_glob = block_n + wtile_n + j * 16 + (lane & 15);
            const int b = n_glob / HoWo;
            const int p = n_glob - b * HoWo;
#pragma unroll
            for (int e = 0; e < 8; ++e) {
                const int m = base_m + e;
                const float vv = acc[i][j][e] * scale[m] + bias[m];
                Out[((size_t)b * Cout + m) * HoWo + p] = vv;
            }
        }
    }
}

// ---------------------------------------------------------------------------
// Elementwise kernels
// ---------------------------------------------------------------------------
__global__ void zero_kernel(float* __restrict__ p, size_t n) {
    size_t i = (size_t)blockIdx.x * blockDim.x + threadIdx.x;
    size_t st = (size_t)gridDim.x * blockDim.x;
    for (; i < n; i += st) p[i] = 0.f;
}

__global__ void fold_kernel(const float* __restrict__ g, const float* __restrict__ be,
                            const float* __restrict__ mu, const float* __restrict__ va,
                            float* __restrict__ s, float* __restrict__ b, int C) {
    int i = blockIdx.x * blockDim.x + threadIdx.x;
    if (i < C) {
        float sv = g[i] * rsqrtf(va[i] + 1e-5f);
        s[i] = sv;
        b[i] = be[i] - mu[i] * sv;
    }
}

// IF neuron layer 1: mem1 += I1; spike; soft reset; accumulate m1; write spikes o1
__global__ void if1_kernel(const float* __restrict__ I1, float* __restrict__ mem1,
                           float* __restrict__ o1, float* __restrict__ m1,
                           const float* __restrict__ vth, int total, int CHW) {
    int i = blockIdx.x * blockDim.x + threadIdx.x;
    if (i >= total) return;
    int cp = i % CHW;
    float th = vth[cp];
    float mv = mem1[i] + I1[i];
    float o  = (mv >= th) ? 1.f : 0.f;
    mem1[i] = mv - o * th;
    o1[i]   = o;
    m1[i]  += o;
}

// IF layers 2 / downsample / output neuron; optionally emit last-step outputs
__global__ void if2_kernel(const float* __restrict__ I2, const float* __restrict__ Id,
                           float* __restrict__ mem2, float* __restrict__ memd,
                           float* __restrict__ memf, float* __restrict__ m3,
                           const float* __restrict__ vth2, const float* __restrict__ vthd,
                           const float* __restrict__ vthf,
                           float* __restrict__ o3_out, float* __restrict__ Iu_out,
                           int last, int total, int CHW) {
    int i = blockIdx.x * blockDim.x + threadIdx.x;
    if (i >= total) return;
    int cp = i % CHW;
    float i2 = I2[i], id = Id[i];

    float t2 = vth2[cp];
    float m2 = mem2[i] + i2;
    float o2 = (m2 >= t2) ? 1.f : 0.f;
    mem2[i] = m2 - o2 * t2;

    float td = vthd[cp];
    float md = memd[i] + id;
    float od = (md >= td) ? 1.f : 0.f;
    memd[i] = md - od * td;

    float outs = i2 + id;
    float tf = vthf[cp];
    float mf = memf[i] + outs;
    float o3 = (mf >= tf) ? 1.f : 0.f;
    memf[i] = mf - o3 * tf;
    m3[i] += o3;

    if (last) { o3_out[i] = o3; Iu_out[i] = outs; }
}

// ANN branch: act = relu(I1c) * (m1 > 0)
__global__ void ann_gate1_kernel(const float* __restrict__ I1c, const float* __restrict__ m1,
                                 float* __restrict__ act, int total) {
    int i = blockIdx.x * blockDim.x + threadIdx.x;
    if (i >= total) return;
    float v = I1c[i];
    v = v > 0.f ? v : 0.f;
    act[i] = (m1[i] > 0.f) ? v : 0.f;
}

// ANN branch: out_c = relu(out2 + identity) * (m3 > 0)
__global__ void ann_gate2_kernel(const float* __restrict__ out2, const float* __restrict__ idc,
                                 const float* __restrict__ m3, float* __restrict__ outc, int total) {
    int i = blockIdx.x * blockDim.x + threadIdx.x;
    if (i >= total) return;
    float v = out2[i] + idc[i];
    v = v > 0.f ? v : 0.f;
    outc[i] = (m3[i] > 0.f) ? v : 0.f;
}

// ---------------------------------------------------------------------------
// Host orchestration
// ---------------------------------------------------------------------------
extern "C" void kernel_launch(void* const* d_in, const int* in_sizes, int n_in,
                              void* d_out, int out_size, void* d_ws, size_t ws_size,
                              hipStream_t stream) {
    const int T = 4, B = 32, Cin = 256, Cout = 512, H = 28, Wd = 28, Ho = 14, Wo = 14;
    const int HoWo = Ho * Wo;
    const int CHW  = Cout * HoWo;                 // 100352
    const int S    = B * CHW;                     // 3,211,264 per feature map
    const int Sin  = B * Cin * H * Wd;            // per-step input size

    const float* inp_s   = (const float*)d_in[0];
    const float* inp_c   = (const float*)d_in[2];
    const float* conv1_w = (const float*)d_in[3];
    const float* conv2_w = (const float*)d_in[4];
    const float* ds_w    = (const float*)d_in[5];
    const float* bn1[4]  = {(const float*)d_in[6],  (const float*)d_in[7],
                            (const float*)d_in[8],  (const float*)d_in[9]};
    const float* bn2[4]  = {(const float*)d_in[10], (const float*)d_in[11],
                            (const float*)d_in[12], (const float*)d_in[13]};
    const float* bnd[4]  = {(const float*)d_in[14], (const float*)d_in[15],
                            (const float*)d_in[16], (const float*)d_in[17]};
    const float* vth1  = (const float*)d_in[18];
    const float* vth2  = (const float*)d_in[19];
    const float* vthds = (const float*)d_in[20];
    const float* vthif = (const float*)d_in[21];

    float* out = (float*)d_out;                   // [o3_last | Iu_last | out_c]

    // workspace layout (floats)
    float* ws   = (float*)d_ws;
    float* mem1 = ws + (size_t)0 * S;
    float* mem2 = ws + (size_t)1 * S;
    float* memd = ws + (size_t)2 * S;
    float* memf = ws + (size_t)3 * S;
    float* m1   = ws + (size_t)4 * S;
    float* m3   = ws + (size_t)5 * S;
    float* o1   = ws + (size_t)6 * S;             // spikes / ANN activation
    float* I1   = ws + (size_t)7 * S;
    float* I2   = ws + (size_t)8 * S;
    float* Id   = ws + (size_t)9 * S;
    float* s1   = ws + (size_t)10 * S;
    float* b1   = s1 + Cout;
    float* s2   = b1 + Cout;
    float* b2   = s2 + Cout;
    float* sd   = b2 + Cout;
    float* bd   = sd + Cout;
    if (ws_size < ((size_t)10 * S + 6 * Cout) * sizeof(float)) return;

    const dim3 cgrid(Cout / BM, (B * HoWo) / BN);  // (4, 49)
    const dim3 cblk(256);
    const int eblk = 256;
    const int egrd = (S + eblk - 1) / eblk;

    // fold BN params -> per-channel scale/bias
    fold_kernel<<<2, 256, 0, stream>>>(bn1[0], bn1[1], bn1[2], bn1[3], s1, b1, Cout);
    fold_kernel<<<2, 256, 0, stream>>>(bn2[0], bn2[1], bn2[2], bn2[3], s2, b2, Cout);
    fold_kernel<<<2, 256, 0, stream>>>(bnd[0], bnd[1], bnd[2], bnd[3], sd, bd, Cout);

    // zero state (mem1,mem2,memd,memf,m1,m3)
    zero_kernel<<<2048, 256, 0, stream>>>(ws, (size_t)6 * S);

    // time loop
    for (int t = 0; t < T; ++t) {
        const float* xt = inp_s + (size_t)t * Sin;
        // conv1: 3x3 s2 p1, 256ch 28x28 -> 512ch 14x14
        conv_wmma_kernel<<<cgrid, cblk, 0, stream>>>(xt, conv1_w, s1, b1, I1,
                                                     Cin, H, Wd, 3, 2, 1, Cout, Ho, Wo, B);
        if1_kernel<<<egrd, eblk, 0, stream>>>(I1, mem1, o1, m1, vth1, S, CHW);
        // conv2: 3x3 s1 p1 on spikes, 512ch 14x14 -> 512ch 14x14
        conv_wmma_kernel<<<cgrid, cblk, 0, stream>>>(o1, conv2_w, s2, b2, I2,
                                                     Cout, Ho, Wo, 3, 1, 1, Cout, Ho, Wo, B);
        // downsample: 1x1 s2 p0
        conv_wmma_kernel<<<cgrid, cblk, 0, stream>>>(xt, ds_w, sd, bd, Id,
                                                     Cin, H, Wd, 1, 2, 0, Cout, Ho, Wo, B);
        if2_kernel<<<egrd, eblk, 0, stream>>>(I2, Id, mem2, memd, memf, m3,
                                              vth2, vthds, vthif,
                                              out, out + S, (t == T - 1) ? 1 : 0, S, CHW);
    }

    // ANN branch (reuses I1/o1/I2/Id scratch)
    conv_wmma_kernel<<<cgrid, cblk, 0, stream>>>(inp_c, conv1_w, s1, b1, I1,
                                                 Cin, H, Wd, 3, 2, 1, Cout, Ho, Wo, B);
    ann_gate1_kernel<<<egrd, eblk, 0, stream>>>(I1, m1, o1, S);
    conv_wmma_kernel<<<cgrid, cblk, 0, stream>>>(o1, conv2_w, s2, b2, I2,
                                                 Cout, Ho, Wo, 3, 1, 1, Cout, Ho, Wo, B);
    conv_wmma_kernel<<<cgrid, cblk, 0, stream>>>(inp_c, ds_w, sd, bd, Id,
                                                 Cin, H, Wd, 1, 2, 0, Cout, Ho, Wo, B);
    ann_gate2_kernel<<<egrd, eblk, 0, stream>>>(I2, Id, m3, out + (size_t)2 * S, S);
}